// LovaszBCEWithBCE_27204322853619
// MI455X (gfx1250) — compile-verified
//
#include <hip/hip_runtime.h>
#include <hip/hip_bf16.h>

#define B_    8
#define C_    16
#define HW_   (512 * 512)                 // 262144 = 2^18
#define PIX_  (B_ * HW_)                  // 2,097,152 pixels
#define TOT_  ((long long)B_ * C_ * HW_)  // 33,554,432 elements
#define NPAIR (B_ * C_)                   // 128 (b,c) pairs
#define BINS  8192
#define MR    8.0f                        // histogram range for m = sign*logit
#define SPLIT 4                           // blocks per (b,c) plane in phase 1
#define EFIX  1048576.0f                  // 2^20 fixed-point scale for err sums
#define CNT_SHIFT 40
#define ERR_MASK  ((1ull << CNT_SHIFT) - 1ull)

typedef unsigned long long u64;
typedef __attribute__((ext_vector_type(2))) float v2f;
typedef __attribute__((ext_vector_type(8))) float v8f;

// ---------------------------------------------------------------- init ----
__global__ void lov_init_kernel(unsigned* ws_u32, int n) {
    int i = blockIdx.x * blockDim.x + threadIdx.x;
    int stride = gridDim.x * blockDim.x;
    for (; i < n; i += stride) ws_u32[i] = 0u;
}

// ----------------------------------------------------------- decode t ----
__global__ __launch_bounds__(256)
void lov_decode_kernel(const long long* __restrict__ target,
                       unsigned char* __restrict__ tcls) {
    int pix = blockIdx.x * blockDim.x + threadIdx.x;
    if (pix < PIX_) {
        long long t = __builtin_nontemporal_load(&target[pix]);  // 0..16
        tcls[pix] = (unsigned char)t;
    }
}

// ------------------------------------------------------------- phase 1 ----
// One block per ((b,c), segment). 64 KB dynamic-LDS histogram of packed
// {count:24 | err_fixpoint:40} u64 bins, DS_ADD_U64 atomics, merged to L2.
// Positives (~6%) bypass LDS and hit global directly. BCE folded in.
__global__ __launch_bounds__(256)
void lov_phase1_kernel(const float* __restrict__ logits,
                       const unsigned char* __restrict__ tcls,
                       u64* __restrict__ pk_all,
                       u64* __restrict__ pk_pos,
                       double* __restrict__ accs) {
    extern __shared__ u64 h_all[];          // [BINS] = 64 KB
    const int q   = blockIdx.x / SPLIT;     // (b,c) pair = plane index
    const int seg = blockIdx.x % SPLIT;
    const int b   = q >> 4;
    const int c   = q & 15;
    const int tid = (int)threadIdx.x;
    const int chunk = HW_ / SPLIT;          // 65536 pixels per block

    for (int j = tid; j < BINS; j += 256) h_all[j] = 0ull;
    __syncthreads();

    const float* lplane = logits + (size_t)q * HW_ + (size_t)seg * chunk;
    const unsigned char* tplane = tcls + (size_t)b * HW_ + (size_t)seg * chunk;
    const float inv_w = (float)BINS / (2.0f * MR);
    const unsigned qb = (unsigned)q * BINS;

    float bce = 0.0f;
    for (int i = tid; i < chunk; i += 256) {
        int   t = (int)tplane[i];
        float l = __builtin_nontemporal_load(&lplane[i]);
        bool  valid = (t < C_);
        bool  oh    = (t == c);
        // ---- BCE: log sigma(l), log(1-sigma(l)), numerically stable
        float lpos = (l >= 0.0f) ? -log1pf(expf(-l)) : (l - log1pf(expf(l)));
        float lneg = lpos - l;
        lpos = fmaxf(lpos, -100.0f);
        lneg = fmaxf(lneg, -100.0f);
        if (valid) bce += oh ? -lpos : -lneg;
        // ---- Lovasz counting-sort histogram: m = s*l, e = 1 - tanh(m)
        bool  y = valid && oh;
        float m = y ? l : -l;
        float e = 1.0f - tanhf(m);
        int bin = (int)((m + MR) * inv_w);
        bin = min(max(bin, 0), BINS - 1);
        u64 packed = (1ull << CNT_SHIFT) | (u64)(unsigned)(e * EFIX + 0.5f);
        atomicAdd(&h_all[bin], packed);                    // DS_ADD_U64
        if (y) atomicAdd(&pk_pos[qb + bin], packed);       // rare -> global
    }
    __syncthreads();
    // merge block histogram into global (skip empty bins)
    for (int j = tid; j < BINS; j += 256) {
        u64 v = h_all[j];
        if (v) atomicAdd(&pk_all[qb + j], v);
    }
    // BCE: wave32 shuffle reduce, one double atomic per wave
    for (int off = 16; off > 0; off >>= 1) bce += __shfl_down(bce, off, 32);
    if ((tid & 31) == 0) atomicAdd(&accs[0], (double)bce);
}

// ------------------------------------------------------------- phase 2 ----
// Inclusive 256-scan via matmul: S = Lstrict*(X*J) + X*U, f32 WMMA 16x16x4.
// Executed by wave 0 only (EXEC all ones). Counts < 2^24 -> exact in f32.
__device__ inline void wmma_scan256(const float* __restrict__ X,
                                    float* __restrict__ S,
                                    float* __restrict__ Wt) {
    int lane = (int)threadIdx.x;      // 0..31
    int half = lane >> 4;             // 0/1
    int mn   = lane & 15;             // row for A, col for B/C/D
    v8f accT = {};                    // X*U  (intra-row inclusive prefix)
    v8f accW = {};                    // X*J  (row sums broadcast)
#pragma unroll
    for (int ks = 0; ks < 4; ++ks) {
        int k0 = ks * 4 + 2 * half;
        v2f a;  a.x = X[mn * 16 + k0];  a.y = X[mn * 16 + k0 + 1];
        v2f bu; bu.x = (k0     <= mn) ? 1.0f : 0.0f;
                bu.y = (k0 + 1 <= mn) ? 1.0f : 0.0f;   // U[k][n] = k <= n
        v2f bj; bj.x = 1.0f; bj.y = 1.0f;              // J = ones
        accT = __builtin_amdgcn_wmma_f32_16x16x4_f32(false, a, false, bu,
                                                     (short)0, accT, false, false);
        accW = __builtin_amdgcn_wmma_f32_16x16x4_f32(false, a, false, bj,
                                                     (short)0, accW, false, false);
    }
#pragma unroll
    for (int vg = 0; vg < 8; ++vg) Wt[(vg + 8 * half) * 16 + mn] = accW[vg];
    // same-wave LDS ops are in-order; reload W1 in B layout
    v8f accS = accT;
#pragma unroll
    for (int ks = 0; ks < 4; ++ks) {
        int k0 = ks * 4 + 2 * half;
        v2f a;  a.x = (k0     < mn) ? 1.0f : 0.0f;     // Lstrict[m][k] = k < m
                a.y = (k0 + 1 < mn) ? 1.0f : 0.0f;
        v2f b;  b.x = Wt[k0 * 16 + mn];  b.y = Wt[(k0 + 1) * 16 + mn];
        accS = __builtin_amdgcn_wmma_f32_16x16x4_f32(false, a, false, b,
                                                     (short)0, accS, false, false);
    }
#pragma unroll
    for (int vg = 0; vg < 8; ++vg) S[(vg + 8 * half) * 16 + mn] = accS[vg];
}

__device__ inline float jacc(float k, float p, float G) {
    if (k <= 0.0f) return 0.0f;
    return 1.0f - (G - p) / (G + k - p);
}

__global__ __launch_bounds__(256)
void lov_phase2_kernel(const u64* __restrict__ pk_all,
                       const u64* __restrict__ pk_pos,
                       double* __restrict__ accs) {
    __shared__ float Xp[256], Xa[256], Sp[256], Sa[256], Wt[256], red[256];
    const int q = blockIdx.x;               // (b,c) pair
    const int t = (int)threadIdx.x;
    const int CH = BINS / 256;              // 32 bins per thread
    const int base = q * BINS + t * CH;
    float sp = 0.0f, sa = 0.0f;
    for (int i = 0; i < CH; ++i) {
        sp += (float)(unsigned)(pk_pos[base + i] >> CNT_SHIFT);
        sa += (float)(unsigned)(pk_all[base + i] >> CNT_SHIFT);
    }
    Xp[t] = sp; Xa[t] = sa;
    __syncthreads();
    if (t < 32) wmma_scan256(Xp, Sp, Wt);
    __syncthreads();
    if (t < 32) wmma_scan256(Xa, Sa, Wt);
    __syncthreads();
    const float G = Sp[255];                // total positives for this (b,c)
    float p = Sp[t] - sp;                   // exclusive prefix of positives
    float k = Sa[t] - sa;                   // exclusive prefix of all
    float loss = 0.0f;
    const float escale = 1.0f / EFIX;
    for (int i = 0; i < CH; ++i) {
        u64 va = pk_all[base + i];
        if (!va) continue;
        u64 vp = pk_pos[base + i];
        float na = (float)(unsigned)(va >> CNT_SHIFT);
        float np = (float)(unsigned)(vp >> CNT_SHIFT);
        float nn = na - np;
        float Ea = (float)(va & ERR_MASK) * escale;
        float Ep = (float)(vp & ERR_MASK) * escale;
        float En = Ea - Ep;
        if (np > 0.0f) {   // positives first within bin (tie-break approx)
            loss += (Ep / np) * (jacc(k + np, p + np, G) - jacc(k, p, G));
            k += np; p += np;
        }
        if (nn > 0.0f) {
            loss += (En / nn) * (jacc(k + nn, p, G) - jacc(k, p, G));
            k += nn;
        }
    }
    red[t] = loss;
    __syncthreads();
    for (int s = 128; s > 0; s >>= 1) {
        if (t < s) red[t] += red[t + s];
        __syncthreads();
    }
    if (t == 0) atomicAdd(&accs[1], (double)red[0]);
}

// ------------------------------------------------------------ finalize ----
__global__ void lov_finalize_kernel(const double* __restrict__ accs,
                                    float* __restrict__ out) {
    // ALPHA=1: mean of 128 per-class lovasz losses + BCE mean over all elems
    out[0] = (float)(accs[1] / (double)NPAIR + accs[0] / (double)TOT_);
}

// --------------------------------------------------------------- launch ---
extern "C" void kernel_launch(void* const* d_in, const int* in_sizes, int n_in,
                              void* d_out, int out_size, void* d_ws, size_t ws_size,
                              hipStream_t stream) {
    const float*     logits = (const float*)d_in[0];
    const long long* target = (const long long*)d_in[1];
    float*           out    = (float*)d_out;

    const size_t arr_elems = (size_t)NPAIR * BINS;          // 1,048,576
    u64*    pk_all = (u64*)d_ws;
    u64*    pk_pos = pk_all + arr_elems;
    double* accs   = (double*)(pk_pos + arr_elems);         // [0]=bce, [1]=lovasz
    unsigned char* tcls = (unsigned char*)(accs + 2);

    const size_t need = 2 * arr_elems * 8 + 2 * sizeof(double) + (size_t)PIX_;
    if (ws_size < need) return;   // workspace too small; nothing sane to do

    const int zero_n = (int)(2 * arr_elems * 2 + 4);        // u32 words to zero
    lov_init_kernel<<<4096, 256, 0, stream>>>((unsigned*)d_ws, zero_n);

    lov_decode_kernel<<<PIX_ / 256, 256, 0, stream>>>(target, tcls);

    lov_phase1_kernel<<<NPAIR * SPLIT, 256, BINS * sizeof(u64), stream>>>(
        logits, tcls, pk_all, pk_pos, accs);

    lov_phase2_kernel<<<NPAIR, 256, 0, stream>>>(pk_all, pk_pos, accs);

    lov_finalize_kernel<<<1, 1, 0, stream>>>(accs, out);
}